// old_Clusterformer_9122510537239
// MI455X (gfx1250) — compile-verified
//
#include <hip/hip_runtime.h>

// Clusterformer local-window branch for MI455X (gfx1250, wave32).
// Bandwidth-bound (~190 MB traffic, ~0.77 GFLOP -> ~8us @ 23.3 TB/s).
// Depthwise conv is block-diagonal over channels (no valid WMMA mapping),
// so the CDNA5 feature used is the async global->LDS DMA path (ASYNCcnt).
// Compute uses sliding-window register reuse: 7-pixel half-rows per wave,
// K+6 LDS loads feed 7*K FMAs per tap row (~3.8x fewer LDS ops per MAC),
// keeping the kernel on the HBM roofline instead of the LDS-issue roofline.

#define HH    8
#define CCH   32
#define HWD   56
#define NPIX  (HWD * HWD)
#define TILE  14   // 56 = 4 * 14 -> 16 tiles per (b,head) plane

template <int K>
__global__ __launch_bounds__(256) void dwconv_mul_kernel(
    const float* __restrict__ q, const float* __restrict__ v,
    const float* __restrict__ w, const float* __restrict__ bias,
    float* __restrict__ out, int head_base)
{
    constexpr int PAD = K / 2;
    constexpr int LW  = TILE + 2 * PAD;      // staged tile edge (16/18/20)
    __shared__ __align__(16) float smem[LW * LW * CCH];

    const int tid   = threadIdx.x;
    const int tx    = blockIdx.x & 3;        // 16 tiles / plane
    const int ty    = blockIdx.x >> 2;
    const int b     = blockIdx.y;            // batch
    const int hloc  = blockIdx.z;            // head within this window group
    const int head  = head_base + hloc;
    const int plane = b * HH + head;

    const float* vp = v + (size_t)plane * NPIX * CCH;
    const float* qp = q + (size_t)plane * NPIX * CCH;

    const int x0 = tx * TILE - PAD;          // image coords of staged tile origin
    const int y0 = ty * TILE - PAD;

    // ---- 1) zero LDS tile (provides the conv zero-padding in the halo) ----
    for (int i = tid * 4; i < LW * LW * CCH; i += 256 * 4) {
        *(float4*)&smem[i] = make_float4(0.f, 0.f, 0.f, 0.f);
    }
    __syncthreads();   // ds stores committed before async-DMA writes can land

    // ---- 2) async global -> LDS copy of in-bounds v pixels (16B chunks) ----
    constexpr int NCHUNK = LW * LW * 8;      // 8 x 16B per pixel (32 ch * fp32)
    for (int i = tid; i < NCHUNK; i += 256) {
        const int chunk = i & 7;
        const int p     = i >> 3;
        const int ly    = p / LW;
        const int lx    = p - ly * LW;
        const int gy    = y0 + ly;
        const int gx    = x0 + lx;
        if ((unsigned)gy < HWD && (unsigned)gx < HWD) {
            unsigned long long ga = (unsigned long long)(const void*)
                (vp + ((size_t)(gy * HWD + gx) * CCH + chunk * 4));
            // flat LDS pointer: low 32 bits are the LDS byte offset
            unsigned la = (unsigned)(unsigned long long)(const void*)
                (&smem[(size_t)(ly * LW + lx) * CCH + chunk * 4]);
            asm volatile("global_load_async_to_lds_b128 %0, %1, off"
                         :: "v"(la), "v"(ga) : "memory");
        }
    }
    asm volatile("s_wait_asynccnt 0" ::: "memory");  // this wave's DMAs done
    __syncthreads();                                 // all waves' DMAs visible

    // ---- 3) per-lane depthwise weights + bias (uniform K per workgroup) ----
    const int c    = tid & (CCH - 1);        // channel owned by this lane
    const int widx = hloc * CCH + c;         // index into this group's w/b arrays
    float wr[K * K];
#pragma unroll
    for (int t = 0; t < K * K; ++t) wr[t] = w[widx * K * K + t];
    const float bv = bias[widx];

    // ---- 4) conv + q-gate; each wave: 7-pixel half-row x 32 channels ----
    // 14x14 tile = 28 half-rows; 8 waves -> 4 iterations (last half-active).
    const int wv = tid >> 5;                 // wave id 0..7
    for (int it = 0; it < 4; ++it) {
        const int r = wv + it * 8;
        if (r < 28) {
            const int py  = r >> 1;
            const int px0 = (r & 1) * 7;

            float acc[7];
#pragma unroll
            for (int i = 0; i < 7; ++i) acc[i] = bv;

#pragma unroll
            for (int dy = 0; dy < K; ++dy) {
                // sliding window: K+6 LDS loads feed 7*K FMAs
                const float* rowp = &smem[((py + dy) * LW + px0) * CCH + c];
                float buf[K + 6];
#pragma unroll
                for (int j = 0; j < K + 6; ++j) buf[j] = rowp[j * CCH];
#pragma unroll
                for (int dx = 0; dx < K; ++dx) {
                    const float wt = wr[dy * K + dx];
#pragma unroll
                    for (int i = 0; i < 7; ++i)
                        acc[i] = fmaf(wt, buf[dx + i], acc[i]);
                }
            }

            const int gy  = ty * TILE + py;
            const int gx0 = tx * TILE + px0;
#pragma unroll
            for (int i = 0; i < 7; ++i) {
                const int n = gy * HWD + gx0 + i;
                const float qv = qp[(size_t)n * CCH + c];
                // out[b, n, head*32 + c]
                out[((size_t)b * NPIX + n) * (HH * CCH) + head * CCH + c]
                    = qv * acc[i];
            }
        }
    }
}

extern "C" void kernel_launch(void* const* d_in, const int* in_sizes, int n_in,
                              void* d_out, int out_size, void* d_ws, size_t ws_size,
                              hipStream_t stream)
{
    (void)in_sizes; (void)n_in; (void)out_size; (void)d_ws; (void)ws_size;
    const float* q  = (const float*)d_in[0];
    const float* v  = (const float*)d_in[1];
    const float* w3 = (const float*)d_in[2];
    const float* b3 = (const float*)d_in[3];
    const float* w5 = (const float*)d_in[4];
    const float* b5 = (const float*)d_in[5];
    const float* w7 = (const float*)d_in[6];
    const float* b7 = (const float*)d_in[7];
    float* out = (float*)d_out;

    // heads 0-1: 3x3 | heads 2-4: 5x5 | heads 5-7: 7x7
    // grid: x = 16 spatial tiles, y = batch (16), z = heads in window group
    dwconv_mul_kernel<3><<<dim3(16, 16, 2), 256, 0, stream>>>(q, v, w3, b3, out, 0);
    dwconv_mul_kernel<5><<<dim3(16, 16, 3), 256, 0, stream>>>(q, v, w5, b5, out, 2);
    dwconv_mul_kernel<7><<<dim3(16, 16, 3), 256, 0, stream>>>(q, v, w7, b7, out, 5);
}